// TGCN2_88759794139277
// MI455X (gfx1250) — compile-verified
//
#include <hip/hip_runtime.h>
#include <math.h>

// ---------------------------------------------------------------------------
// TGCN cell: 2x (GCNConv+ReLU) branches (z and h; r is dead code), mean pool,
// GRU-style gate combine.  fp32 WMMA (v_wmma_f32_16x16x4_f32) for the GEMMs.
// GCN epilogue (self-loop + bias + relu) is fused into GEMM2's LDS staging.
// ---------------------------------------------------------------------------

typedef float v2f __attribute__((ext_vector_type(2)));
typedef float v8f __attribute__((ext_vector_type(8)));

// ---------------- WMMA GEMM core (A tile already staged in LDS) -------------
template<int K, int NCOL>
__device__ __forceinline__ void wmma_tile_compute(const float* sA,
                                                  const float* __restrict__ W,
                                                  float* __restrict__ C,
                                                  int rowBase, int tid) {
  const int wave = tid >> 5;
  const int lane = tid & 31;
  const int colBase = wave * 16;
  const int lhalf = lane >> 4;   // 0: K+{0,1}, 1: K+{2,3}  (fp32 frag layout)
  const int lmod = lane & 15;

  v8f acc = {};
#pragma unroll
  for (int kb = 0; kb < K; kb += 4) {
    const int k0 = kb + 2 * lhalf;
    v2f a, b;
    a.x = sA[lmod * K + k0];
    a.y = sA[lmod * K + k0 + 1];
    b.x = W[(size_t)k0 * NCOL + colBase + lmod];
    b.y = W[(size_t)(k0 + 1) * NCOL + colBase + lmod];
    acc = __builtin_amdgcn_wmma_f32_16x16x4_f32(
        /*neg_a=*/false, a, /*neg_b=*/false, b,
        /*c_mod=*/(short)0, acc, /*reuse_a=*/false, /*reuse_b=*/false);
  }

  // C/D layout: VGPR r -> row r (lanes 0-15) / row r+8 (lanes 16-31)
#pragma unroll
  for (int r = 0; r < 8; ++r) {
    const int row = rowBase + r + 8 * lhalf;
    C[(size_t)row * NCOL + colBase + lmod] = acc[r];
  }
}

// ---------------- GEMM1: C = A @ W ------------------------------------------
template<int K, int NCOL>
__global__ __launch_bounds__((NCOL / 16) * 32)
void gemm_wmma_kernel(const float* __restrict__ A, const float* __restrict__ W,
                      float* __restrict__ C, int nrows) {
  __shared__ float sA[16 * K];
  const int rowBase = blockIdx.x * 16;
  const int tid = threadIdx.x;
  constexpr int NTHREADS = (NCOL / 16) * 32;
  for (int i = tid; i < 16 * K; i += NTHREADS) {
    const int r = i / K;
    const int k = i - r * K;
    sA[i] = A[(size_t)(rowBase + r) * K + k];
  }
  __syncthreads();
  wmma_tile_compute<K, NCOL>(sA, W, C, rowBase, tid);
}

// ---------------- GEMM2 with fused GCN epilogue on A ------------------------
// A = relu(scat + dinv^2 * xw + bias), computed on the fly while staging LDS.
template<int K, int NCOL>
__global__ __launch_bounds__((NCOL / 16) * 32)
void gemm_wmma_fused_relu_kernel(const float* __restrict__ scat,
                                 const float* __restrict__ xw,
                                 const float* __restrict__ dinv,
                                 const float* __restrict__ bias,
                                 const float* __restrict__ W,
                                 float* __restrict__ C, int nrows) {
  __shared__ float sA[16 * K];
  const int rowBase = blockIdx.x * 16;
  const int tid = threadIdx.x;
  constexpr int NTHREADS = (NCOL / 16) * 32;
  for (int i = tid; i < 16 * K; i += NTHREADS) {
    const int r = i / K;
    const int k = i - r * K;
    const size_t idx = (size_t)(rowBase + r) * K + k;
    const float di = dinv[rowBase + r];
    float v = scat[idx] + di * di * xw[idx] + bias[k];
    sA[i] = v > 0.f ? v : 0.f;
  }
  __syncthreads();
  wmma_tile_compute<K, NCOL>(sA, W, C, rowBase, tid);
}

// ---------------- degree / norm ---------------------------------------------
__global__ __launch_bounds__(256)
void degree_kernel(const int* __restrict__ dst, const float* __restrict__ ew,
                   float* __restrict__ deg, int nedges) {
  int e = blockIdx.x * 256 + threadIdx.x;
  if (e < nedges) atomicAdd(&deg[dst[e]], ew[e]);
}

__global__ __launch_bounds__(256)
void dinv_kernel(float* __restrict__ deg, int n) {
  int i = blockIdx.x * 256 + threadIdx.x;
  if (i < n) {
    float d = deg[i] + 1.0f;  // self-loop weight 1 on every node
    deg[i] = rsqrtf(fmaxf(d, 1e-30f));
  }
}

__global__ __launch_bounds__(256)
void counts_kernel(const int* __restrict__ batch, float* __restrict__ counts, int n) {
  int i = blockIdx.x * 256 + threadIdx.x;
  if (i < n) atomicAdd(&counts[batch[i]], 1.0f);
}

// ---------------- edge scatter: out[dst] += norm * xw[src] ------------------
template<int NCOL>
__global__ __launch_bounds__(256)
void scatter_kernel(const float* __restrict__ XW, const int* __restrict__ src,
                    const int* __restrict__ dst, const float* __restrict__ ew,
                    const float* __restrict__ dinv, float* __restrict__ out,
                    int nedges) {
  constexpr int LPE = NCOL / 4;  // lanes per edge, float4 per lane
  long long t = (long long)blockIdx.x * 256 + threadIdx.x;
  int e = (int)(t / LPE);
  int c = (int)(t % LPE) * 4;
  if (e >= nedges) return;
  const int s = src[e];
  const int d = dst[e];
  const float nrm = dinv[s] * ew[e] * dinv[d];
  const float4 x = *(const float4*)(XW + (size_t)s * NCOL + c);
  float* o = out + (size_t)d * NCOL + c;
  atomicAdd(o + 0, nrm * x.x);
  atomicAdd(o + 1, nrm * x.y);
  atomicAdd(o + 2, nrm * x.z);
  atomicAdd(o + 3, nrm * x.w);
}

// ---------------- self-loop + bias + relu + segment-sum pool ----------------
__global__ __launch_bounds__(256)
void pool_kernel(const float* __restrict__ acc, const float* __restrict__ XW,
                 const float* __restrict__ dinv, const float* __restrict__ bias,
                 const int* __restrict__ batch, float* __restrict__ pooled, int n) {
  __shared__ float sp[64 * 64];
  for (int i = threadIdx.x; i < 64 * 64; i += 256) sp[i] = 0.f;
  __syncthreads();
  const int row = blockIdx.x * 256 + threadIdx.x;
  if (row < n) {
    const int b = batch[row];
    const float di = dinv[row];
    const float d2 = di * di;
    const float* ar = acc + (size_t)row * 64;
    const float* xr = XW + (size_t)row * 64;
#pragma unroll 4
    for (int c = 0; c < 64; ++c) {
      float v = ar[c] + d2 * xr[c] + bias[c];
      v = v > 0.f ? v : 0.f;
      atomicAdd(&sp[b * 64 + c], v);  // ds_add_f32
    }
  }
  __syncthreads();
  for (int i = threadIdx.x; i < 64 * 64; i += 256) {
    const float v = sp[i];
    if (v != 0.f) atomicAdd(&pooled[i], v);
  }
}

// ---------------- final GRU-style gate combine ------------------------------
__global__ __launch_bounds__(256)
void gate_kernel(const float* __restrict__ pooled0, const float* __restrict__ pooled2,
                 const float* __restrict__ counts, const float* __restrict__ H,
                 const float* __restrict__ Wl, const float* __restrict__ bl,
                 float* __restrict__ out) {
  const int tid = blockIdx.x * 256 + threadIdx.x;  // 4096 = 64 graphs x 64 feat
  const int b = tid >> 6;
  const int j = tid & 63;
  const float inv = 1.0f / fmaxf(counts[b], 1.0f);
  const float* Wl0 = Wl;                 // gate z: [128,64]
  const float* Wl2 = Wl + 2 * 128 * 64;  // gate h
  float z = bl[j];
  float ht = bl[2 * 64 + j];
  for (int k = 0; k < 64; ++k) {
    const float p0 = pooled0[b * 64 + k] * inv;
    const float p2 = pooled2[b * 64 + k] * inv;
    z += p0 * Wl0[k * 64 + j];
    ht += p2 * Wl2[k * 64 + j];
  }
  for (int k = 0; k < 64; ++k) {
    const float h = H[b * 64 + k];
    z += h * Wl0[(64 + k) * 64 + j];
    ht += h * Wl2[(64 + k) * 64 + j];
  }
  const float zs = 1.0f / (1.0f + expf(-z));
  const float hts = tanhf(ht);
  out[tid] = zs * H[tid] + (1.0f - zs) * hts;
}

// ---------------------------------------------------------------------------
extern "C" void kernel_launch(void* const* d_in, const int* in_sizes, int n_in,
                              void* d_out, int out_size, void* d_ws, size_t ws_size,
                              hipStream_t stream) {
  (void)n_in; (void)out_size; (void)ws_size;
  const float* X     = (const float*)d_in[0];
  const int*   ei    = (const int*)d_in[1];   // [2,E]
  const int*   batch = (const int*)d_in[2];
  const float* ew    = (const float*)d_in[3];
  const float* H     = (const float*)d_in[4];
  const float* W1    = (const float*)d_in[5];  // [3,64,128]
  const float* b1    = (const float*)d_in[6];  // [3,128]
  const float* W2    = (const float*)d_in[7];  // [3,128,64]
  const float* b2    = (const float*)d_in[8];  // [3,64]
  const float* Wl    = (const float*)d_in[9];  // [3,128,64]
  const float* bl    = (const float*)d_in[10]; // [3,64]

  const int N = in_sizes[0] / 64;   // 100000 (divisible by 16)
  const int E = in_sizes[3];        // 800000
  const int* src = ei;
  const int* dst = ei + E;

  char* ws = (char*)d_ws;
  size_t off = 0;
  auto take = [&](size_t bytes) -> char* {
    char* p = ws + off;
    off = (off + bytes + 255) & ~(size_t)255;
    return p;
  };
  float* dinv    = (float*)take((size_t)N * 4);
  float* counts  = (float*)take(64 * 4);
  float* pooled0 = (float*)take(64 * 64 * 4);
  float* pooled2 = (float*)take(64 * 64 * 4);
  float* xw1     = (float*)take((size_t)N * 128 * 4);
  float* scat1   = (float*)take((size_t)N * 128 * 4);
  float* xw2     = (float*)take((size_t)N * 64 * 4);
  float* scat2   = (float*)take((size_t)N * 64 * 4);

  hipMemsetAsync(dinv, 0, (size_t)N * 4, stream);
  hipMemsetAsync(counts, 0, 64 * 4, stream);
  hipMemsetAsync(pooled0, 0, 64 * 64 * 4, stream);
  hipMemsetAsync(pooled2, 0, 64 * 64 * 4, stream);

  degree_kernel<<<(E + 255) / 256, 256, 0, stream>>>(dst, ew, dinv, E);
  dinv_kernel<<<(N + 255) / 256, 256, 0, stream>>>(dinv, N);
  counts_kernel<<<(N + 255) / 256, 256, 0, stream>>>(batch, counts, N);

  const int rowTiles = (N + 15) / 16;
  for (int gi = 0; gi < 2; ++gi) {
    const int g = (gi == 0) ? 0 : 2;            // branch 1 (reset gate) is dead
    float* pooled = (gi == 0) ? pooled0 : pooled2;

    // layer 1: xw1 = X @ W1[g]
    gemm_wmma_kernel<64, 128><<<rowTiles, 256, 0, stream>>>(
        X, W1 + (size_t)g * 64 * 128, xw1, N);
    hipMemsetAsync(scat1, 0, (size_t)N * 128 * 4, stream);
    scatter_kernel<128><<<(int)(((long long)E * 32 + 255) / 256), 256, 0, stream>>>(
        xw1, src, dst, ew, dinv, scat1, E);

    // layer 2: xw2 = relu(scat1 + dinv^2*xw1 + b1[g]) @ W2[g]  (epilogue fused)
    gemm_wmma_fused_relu_kernel<128, 64><<<rowTiles, 128, 0, stream>>>(
        scat1, xw1, dinv, b1 + g * 128, W2 + (size_t)g * 128 * 64, xw2, N);
    hipMemsetAsync(scat2, 0, (size_t)N * 64 * 4, stream);
    scatter_kernel<64><<<(int)(((long long)E * 16 + 255) / 256), 256, 0, stream>>>(
        xw2, src, dst, ew, dinv, scat2, E);
    pool_kernel<<<(N + 255) / 256, 256, 0, stream>>>(
        scat2, xw2, dinv, b2 + g * 64, batch, pooled, N);
  }

  gate_kernel<<<16, 256, 0, stream>>>(pooled0, pooled2, counts, H, Wl, bl,
                                      (float*)d_out);
}